// FlexibleTripletSampler_48730698940865
// MI455X (gfx1250) — compile-verified
//
#include <hip/hip_runtime.h>

// ---------------------------------------------------------------------------
// Fused triplet miner for MI455X (gfx1250, wave32, WMMA f32 16x16x4).
//   k1: row norms
//   k2: WMMA-tiled Gram + fused masked argmins, double-buffered ASYNC
//       global->LDS staging (global_load_async_to_lds_b128 / ASYNCcnt)
//   k3: merge column-split partials (+ empty-positive fallback)
//   k4: gather outputs (a_idx, x, x[p], x[n])
// ---------------------------------------------------------------------------

typedef float v2f __attribute__((ext_vector_type(2)));
typedef float v8f __attribute__((ext_vector_type(8)));

#define N_PTS 8192
#define DIM 128
#define NTILES (N_PTS / 16)      // 512 column tiles of 16
#define SPLIT 4                  // column-space split for occupancy
#define TILES_PER_SPLIT (NTILES / SPLIT)
#define WAVES_PER_BLOCK 8
#define LDS_STRIDE 132           // 128 + 4 pad -> conflict-free ds_load_b64
#define EPS_SQ 1e-4f
#define FLT_BIG 3.0e38f

// Async global->LDS B128 copy (CDNA5 path, tracked by ASYNCcnt).
// VDST VGPR supplies the LDS byte address (flat addr low 32 bits == LDS offset).
__device__ __forceinline__ void async_copy_b128(const float* gsrc, float* lds_dst) {
  unsigned loff = (unsigned)(size_t)lds_dst;
  unsigned long long gaddr = (unsigned long long)(size_t)gsrc;
  asm volatile("global_load_async_to_lds_b128 %0, %1, off"
               :: "v"(loff), "v"(gaddr)
               : "memory");
}

__device__ __forceinline__ void wait_async0() {
#if __has_builtin(__builtin_amdgcn_s_wait_asynccnt)
  __builtin_amdgcn_s_wait_asynccnt(0);
#else
  asm volatile("s_wait_asynccnt 0x0" ::: "memory");
#endif
}

// ---------------- k1: row squared norms ------------------------------------
__global__ __launch_bounds__(256) void norms_kernel(const float* __restrict__ x,
                                                    float* __restrict__ nrm) {
  int row = blockIdx.x * blockDim.x + threadIdx.x;
  if (row < N_PTS) {
    const float4* xr = (const float4*)(x + (size_t)row * DIM);
    float s = 0.f;
#pragma unroll
    for (int k = 0; k < DIM / 4; ++k) {
      float4 v = xr[k];
      s += v.x * v.x + v.y * v.y + v.z * v.z + v.w * v.w;
    }
    nrm[row] = s;
  }
}

// ---------------- k2: fused WMMA Gram + masked argmins ----------------------
// grid.x = (N_PTS/16/WAVES_PER_BLOCK) * SPLIT ; block = 256 (8 waves)
// Each wave owns one 16-row tile; block shares double-buffered LDS B tiles.
__global__ __launch_bounds__(256) void triplet_argmin_kernel(
    const float* __restrict__ x, const int* __restrict__ labels,
    const float* __restrict__ nrm,
    float* __restrict__ pV, int* __restrict__ pI,
    float* __restrict__ nV, int* __restrict__ nI,
    float* __restrict__ aV, int* __restrict__ aI) {
  __shared__ float Bs[2][16 * LDS_STRIDE];

  const int tid  = threadIdx.x;
  const int wave = tid >> 5;
  const int lane = tid & 31;
  const int lo   = lane & 15;   // column within tile (B/C/D N index)
  const int hi   = lane >> 4;   // half-wave selector

  const int rg = blockIdx.x / SPLIT;        // row group (128 rows)
  const int sp = blockIdx.x % SPLIT;        // column split
  const int i0 = (rg * WAVES_PER_BLOCK + wave) * 16;

  // Per-thread assignment for cooperative tile staging: 2 x float4 each.
  const int q0  = tid * 2;
  const int sr0 = (q0 + 0) >> 5, sc0 = (q0 + 0) & 31;  // row / float4-col
  const int sr1 = (q0 + 1) >> 5, sc1 = (q0 + 1) & 31;

  // Preload A fragments for this wave's 16-row tile: 32 k-steps x v2f.
  // f32 WMMA A layout: lane<16 -> A[lo][4k+0..1], lane>=16 -> A[lo][4k+2..3].
  v2f afrag[32];
  const float* arow = x + (size_t)(i0 + lo) * DIM + hi * 2;
#pragma unroll
  for (int k = 0; k < 32; ++k)
    afrag[k] = *(const v2f*)(arow + k * 4);

  // Per-lane row metadata (C/D layout: VGPR r -> M = r + 8*hi).
  float nrmI[8];
  int labI[8];
#pragma unroll
  for (int r = 0; r < 8; ++r) {
    int ri = i0 + r + hi * 8;
    nrmI[r] = nrm[ri];
    labI[r] = labels[ri];
  }

  float bpV[8], bnV[8], baV[8];
  int bpI[8], bnI[8], baI[8];
#pragma unroll
  for (int r = 0; r < 8; ++r) {
    bpV[r] = bnV[r] = baV[r] = FLT_BIG;
    bpI[r] = bnI[r] = baI[r] = 0x7fffffff;
  }

  const int jbeg = sp * TILES_PER_SPLIT;
  const int jend = jbeg + TILES_PER_SPLIT;

  // Prologue: async-stage first tile into buffer 0.
  {
    const int j0 = jbeg * 16;
    async_copy_b128(x + (size_t)(j0 + sr0) * DIM + sc0 * 4, &Bs[0][sr0 * LDS_STRIDE + sc0 * 4]);
    async_copy_b128(x + (size_t)(j0 + sr1) * DIM + sc1 * 4, &Bs[0][sr1 * LDS_STRIDE + sc1 * 4]);
  }
  wait_async0();
  __syncthreads();

  for (int jt = jbeg; jt < jend; ++jt) {
    const int j0  = jt * 16;
    const int buf = (jt - jbeg) & 1;

    // Prefetch next tile into the other buffer (overlaps with WMMA phase).
    if (jt + 1 < jend) {
      const int jn = (jt + 1) * 16;
      async_copy_b128(x + (size_t)(jn + sr0) * DIM + sc0 * 4,
                      &Bs[buf ^ 1][sr0 * LDS_STRIDE + sc0 * 4]);
      async_copy_b128(x + (size_t)(jn + sr1) * DIM + sc1 * 4,
                      &Bs[buf ^ 1][sr1 * LDS_STRIDE + sc1 * 4]);
    }

    // 16x16 Gram tile: 32 x V_WMMA_F32_16X16X4_F32 accumulation.
    v8f c = {0.f, 0.f, 0.f, 0.f, 0.f, 0.f, 0.f, 0.f};
    const float* brow = &Bs[buf][lo * LDS_STRIDE + hi * 2];
#pragma unroll
    for (int k = 0; k < 32; ++k) {
      v2f b = *(const v2f*)(brow + k * 4);
      c = __builtin_amdgcn_wmma_f32_16x16x4_f32(false, afrag[k], false, b,
                                                (short)0, c, false, false);
    }

    // Fused epilogue: squared distance + masked argmin updates.
    // Self-exclusion only possible in the (single) diagonal tile (j0 == i0),
    // since row/col tiles are both 16-aligned -> uniform branch.
    const int jcol   = j0 + lo;
    const float nrmJ = nrm[jcol];
    const int labJ   = labels[jcol];
    if (j0 == i0) {
#pragma unroll
      for (int r = 0; r < 8; ++r) {
        float sq  = fmaxf(nrmI[r] + nrmJ - 2.0f * c[r], 0.0f);
        float cmp = fmaxf(sq, EPS_SQ);
        bool same = (labI[r] == labJ);
        bool self = ((r + hi * 8) == lo);
        if (cmp < baV[r])                  { baV[r] = cmp; baI[r] = jcol; }
        if (same && !self && cmp < bpV[r]) { bpV[r] = cmp; bpI[r] = jcol; }
        if (!same && cmp < bnV[r])         { bnV[r] = cmp; bnI[r] = jcol; }
      }
    } else {
#pragma unroll
      for (int r = 0; r < 8; ++r) {
        float sq  = fmaxf(nrmI[r] + nrmJ - 2.0f * c[r], 0.0f);
        float cmp = fmaxf(sq, EPS_SQ);
        bool same = (labI[r] == labJ);
        if (cmp < baV[r])          { baV[r] = cmp; baI[r] = jcol; }
        if (same && cmp < bpV[r])  { bpV[r] = cmp; bpI[r] = jcol; }
        if (!same && cmp < bnV[r]) { bnV[r] = cmp; bnI[r] = jcol; }
      }
    }

    // Next tile's async fill must be complete before anyone reads buf^1,
    // and everyone must be done reading buf before it is refilled next iter.
    wait_async0();
    __syncthreads();
  }

  // Cross-lane merge within each 16-lane half (first-index tie-break).
#pragma unroll
  for (int r = 0; r < 8; ++r) {
    float pv = bpV[r], nv = bnV[r], av = baV[r];
    int pi = bpI[r], ni = bnI[r], ai = baI[r];
#pragma unroll
    for (int m = 1; m < 16; m <<= 1) {
      float ov; int oi;
      ov = __shfl_xor(pv, m, 32); oi = __shfl_xor(pi, m, 32);
      if (ov < pv || (ov == pv && oi < pi)) { pv = ov; pi = oi; }
      ov = __shfl_xor(nv, m, 32); oi = __shfl_xor(ni, m, 32);
      if (ov < nv || (ov == nv && oi < ni)) { nv = ov; ni = oi; }
      ov = __shfl_xor(av, m, 32); oi = __shfl_xor(ai, m, 32);
      if (ov < av || (ov == av && oi < ai)) { av = ov; ai = oi; }
    }
    if (lo == 0) {
      int row = i0 + r + hi * 8;
      size_t q = (size_t)row * SPLIT + sp;
      pV[q] = pv; pI[q] = pi;
      nV[q] = nv; nI[q] = ni;
      aV[q] = av; aI[q] = ai;
    }
  }
}

// ---------------- k3: merge split partials ----------------------------------
__global__ __launch_bounds__(256) void merge_kernel(
    const float* __restrict__ pV, const int* __restrict__ pI,
    const float* __restrict__ nV, const int* __restrict__ nI,
    const float* __restrict__ aV, const int* __restrict__ aI,
    int* __restrict__ pidx, int* __restrict__ nidx) {
  int row = blockIdx.x * blockDim.x + threadIdx.x;
  if (row >= N_PTS) return;
  float bp = FLT_BIG, bn = FLT_BIG, ba = FLT_BIG;
  int bpi = 0x7fffffff, bni = 0x7fffffff, bai = 0x7fffffff;
#pragma unroll
  for (int s = 0; s < SPLIT; ++s) {
    size_t q = (size_t)row * SPLIT + s;
    float v; int i;
    v = pV[q]; i = pI[q];
    if (v < bp || (v == bp && i < bpi)) { bp = v; bpi = i; }
    v = nV[q]; i = nI[q];
    if (v < bn || (v == bn && i < bni)) { bn = v; bni = i; }
    v = aV[q]; i = aI[q];
    if (v < ba || (v == ba && i < bai)) { ba = v; bai = i; }
  }
  // No positive exists -> reference's uniform "+big" reduces to global argmin.
  pidx[row] = (bp >= FLT_BIG) ? bai : bpi;
  nidx[row] = bni;
}

// ---------------- k4: gather outputs ----------------------------------------
// d_out = [a_idx (n floats) | x (n*d) | x[p_idx] (n*d) | x[n_idx] (n*d)]
__global__ __launch_bounds__(128) void gather_out_kernel(
    const float* __restrict__ x, const int* __restrict__ pidx,
    const int* __restrict__ nidx, float* __restrict__ out) {
  const int i = blockIdx.x;
  const int t = threadIdx.x;
  if (t == 0) out[i] = (float)i;
  float* oa = out + N_PTS;
  float* op = oa + (size_t)N_PTS * DIM;
  float* on = op + (size_t)N_PTS * DIM;
  oa[(size_t)i * DIM + t] = x[(size_t)i * DIM + t];
  int p = pidx[i], g = nidx[i];
  op[(size_t)i * DIM + t] = x[(size_t)p * DIM + t];
  on[(size_t)i * DIM + t] = x[(size_t)g * DIM + t];
}

// ---------------------------------------------------------------------------
extern "C" void kernel_launch(void* const* d_in, const int* in_sizes, int n_in,
                              void* d_out, int out_size, void* d_ws, size_t ws_size,
                              hipStream_t stream) {
  const float* x      = (const float*)d_in[0];
  const int*   labels = (const int*)d_in[1];
  float* out = (float*)d_out;

  // Workspace layout.
  char* ws = (char*)d_ws;
  const size_t PAR = (size_t)N_PTS * SPLIT;
  float* pV = (float*)(ws);                       ws += PAR * sizeof(float);
  int*   pI = (int*)  (ws);                       ws += PAR * sizeof(int);
  float* nV = (float*)(ws);                       ws += PAR * sizeof(float);
  int*   nI = (int*)  (ws);                       ws += PAR * sizeof(int);
  float* aV = (float*)(ws);                       ws += PAR * sizeof(float);
  int*   aI = (int*)  (ws);                       ws += PAR * sizeof(int);
  int*   pidx = (int*)(ws);                       ws += N_PTS * sizeof(int);
  int*   nidx = (int*)(ws);                       ws += N_PTS * sizeof(int);
  float* nrm  = (float*)(ws);

  norms_kernel<<<(N_PTS + 255) / 256, 256, 0, stream>>>(x, nrm);

  const int row_groups = N_PTS / 16 / WAVES_PER_BLOCK;  // 64
  triplet_argmin_kernel<<<row_groups * SPLIT, 256, 0, stream>>>(
      x, labels, nrm, pV, pI, nV, nI, aV, aI);

  merge_kernel<<<(N_PTS + 255) / 256, 256, 0, stream>>>(
      pV, pI, nV, nI, aV, aI, pidx, nidx);

  gather_out_kernel<<<N_PTS, 128, 0, stream>>>(x, pidx, nidx, out);
}